// ALIGNNLayer_36593121362364
// MI455X (gfx1250) — compile-verified
//
#include <hip/hip_runtime.h>

typedef float v2f __attribute__((ext_vector_type(2)));
typedef float v8f __attribute__((ext_vector_type(8)));

#define DFEAT 128

// ---------------------------------------------------------------------------
// GEMM: C[M,128] = A[M,128] @ W[128,128] (+bias) (+relu), fp32 WMMA 16x16x4.
// 128 threads = 4 waves; each wave computes a 16x128 output stripe.
// W staged into LDS (64 KB). Safe for in-place use (C == A): each wave reads
// only the 16 rows it later writes, and all reads feed the accumulators the
// stores depend on.
// ---------------------------------------------------------------------------
__global__ __launch_bounds__(128) void gemm128_wmma(
    const float* A, const float* __restrict__ W,
    const float* __restrict__ bias, float* C, int M, int act)
{
    __shared__ float sW[DFEAT * DFEAT];
    const int tid = threadIdx.x;

    // cooperative load of W into LDS: 16384 floats / 128 threads = 32 float4 each
    {
        const float4* W4 = (const float4*)W;
        float4* sW4 = (float4*)sW;
        #pragma unroll
        for (int i = 0; i < (DFEAT * DFEAT / 4) / 128; ++i)
            sW4[tid + i * 128] = W4[tid + i * 128];
    }
    __syncthreads();

    const int wave = tid >> 5;
    const int lane = tid & 31;
    const int tile = blockIdx.x * 4 + wave;
    const int row0 = tile * 16;
    if (row0 >= M) return;              // wave-uniform; EXEC stays all-ones

    const int half = lane >> 4;         // 0: K={0,1}, 1: K={2,3}
    const int l    = lane & 15;

    v8f zero = {};
    v8f acc[8];
    #pragma unroll
    for (int t = 0; t < 8; ++t) acc[t] = zero;

    // A fragment source: lane l holds row (row0+l), K-pair (k + half*2)
    const float* Arow = A + (size_t)(row0 + l) * DFEAT + half * 2;

    for (int k = 0; k < DFEAT; k += 4) {
        v2f a;
        a[0] = Arow[k];
        a[1] = Arow[k + 1];
        const int kb = k + half * 2;
        #pragma unroll
        for (int t = 0; t < 8; ++t) {
            const int n = t * 16 + l;
            v2f b;
            b[0] = sW[kb * DFEAT + n];
            b[1] = sW[(kb + 1) * DFEAT + n];
            acc[t] = __builtin_amdgcn_wmma_f32_16x16x4_f32(
                false, a, false, b, (short)0, acc[t], false, false);
        }
    }

    // D layout: VGPR r -> row (row0 + r + half*8), col = t*16 + l
    const int rbase = row0 + half * 8;
    #pragma unroll
    for (int t = 0; t < 8; ++t) {
        const int n = t * 16 + l;
        const float bn = bias ? bias[n] : 0.0f;
        #pragma unroll
        for (int r = 0; r < 8; ++r) {
            float v = acc[t][r] + bn;
            if (act) v = fmaxf(v, 0.0f);
            C[(size_t)(rbase + r) * DFEAT + n] = v;
        }
    }
}

// ---------------------------------------------------------------------------
// Elementwise / scatter helpers
// ---------------------------------------------------------------------------
__global__ void set_const_f32(float* p, float v, int n) {
    int i = blockIdx.x * blockDim.x + threadIdx.x;
    if (i < n) p[i] = v;
}

__global__ void deg_count(const long long* __restrict__ dst,
                          float* __restrict__ deg, int nE) {
    int e = blockIdx.x * blockDim.x + threadIdx.x;
    if (e < nE) atomicAdd(&deg[(int)dst[e]], 1.0f);
}

__global__ void rsqrt_inplace(float* p, int n) {
    int i = blockIdx.x * blockDim.x + threadIdx.x;
    if (i < n) p[i] = rsqrtf(p[i]);   // deg >= 1 always (self loop)
}

// out[i] = xw[i] * dinv[i]^2   (self-loop contribution)
__global__ void gcn_self_init(const float* __restrict__ xw,
                              const float* __restrict__ dinv,
                              float* __restrict__ out, int M) {
    long long gid = blockIdx.x * (long long)blockDim.x + threadIdx.x;
    long long i = gid >> 5;
    int c = (int)(gid & 31);
    if (i >= M) return;
    float s = dinv[i]; s = s * s;
    float4 v = ((const float4*)(xw + i * DFEAT))[c];
    v.x *= s; v.y *= s; v.z *= s; v.w *= s;
    ((float4*)(out + i * DFEAT))[c] = v;
}

// out[dst] += xw[src] * dinv[src]*dinv[dst]   (one wave per edge, float4 chunks)
__global__ void gcn_scatter(const float* __restrict__ xw,
                            const long long* __restrict__ src,
                            const long long* __restrict__ dst,
                            const float* __restrict__ dinv,
                            float* out, int nE) {
    long long gid = blockIdx.x * (long long)blockDim.x + threadIdx.x;
    long long e = gid >> 5;
    int c = (int)(gid & 31);
    if (e >= nE) return;
    int s = (int)src[e], d = (int)dst[e];
    float norm = dinv[s] * dinv[d];
    float4 v = ((const float4*)(xw + (size_t)s * DFEAT))[c];
    float* o = out + (size_t)d * DFEAT + c * 4;
    atomicAdd(o + 0, v.x * norm);
    atomicAdd(o + 1, v.y * norm);
    atomicAdd(o + 2, v.z * norm);
    atomicAdd(o + 3, v.w * norm);
}

// out = relu(out + bias)
__global__ void bias_relu(float* out, const float* __restrict__ bias, int M) {
    long long gid = blockIdx.x * (long long)blockDim.x + threadIdx.x;
    long long i = gid >> 5;
    int c = (int)(gid & 31);
    if (i >= M) return;
    float4 b = ((const float4*)bias)[c];
    float4* p = (float4*)(out + i * DFEAT) + c;
    float4 v = *p;
    v.x = fmaxf(v.x + b.x, 0.0f);
    v.y = fmaxf(v.y + b.y, 0.0f);
    v.z = fmaxf(v.z + b.z, 0.0f);
    v.w = fmaxf(v.w + b.w, 0.0f);
    *p = v;
}

// bond[e] = (xa[row[e]] + xa[col[e]]) * 0.5
__global__ void bond_mean(const float* __restrict__ xa,
                          const long long* __restrict__ row,
                          const long long* __restrict__ col,
                          float* __restrict__ out, int nE) {
    long long gid = blockIdx.x * (long long)blockDim.x + threadIdx.x;
    long long e = gid >> 5;
    int c = (int)(gid & 31);
    if (e >= nE) return;
    int r = (int)row[e], cl = (int)col[e];
    float4 va = ((const float4*)(xa + (size_t)r  * DFEAT))[c];
    float4 vb = ((const float4*)(xa + (size_t)cl * DFEAT))[c];
    float4 o;
    o.x = (va.x + vb.x) * 0.5f;
    o.y = (va.y + vb.y) * 0.5f;
    o.z = (va.z + vb.z) * 0.5f;
    o.w = (va.w + vb.w) * 0.5f;
    ((float4*)(out + e * DFEAT))[c] = o;
}

// xa[row[e]] += b2a[e]; xa[col[e]] += b2a[e]
__global__ void b2a_scatter(const float* __restrict__ b2a,
                            const long long* __restrict__ row,
                            const long long* __restrict__ col,
                            float* xa, int nE) {
    long long gid = blockIdx.x * (long long)blockDim.x + threadIdx.x;
    long long e = gid >> 5;
    int c = (int)(gid & 31);
    if (e >= nE) return;
    float4 v = ((const float4*)(b2a + e * DFEAT))[c];
    int r = (int)row[e], cl = (int)col[e];
    float* pr = xa + (size_t)r  * DFEAT + c * 4;
    float* pc = xa + (size_t)cl * DFEAT + c * 4;
    atomicAdd(pr + 0, v.x); atomicAdd(pr + 1, v.y);
    atomicAdd(pr + 2, v.z); atomicAdd(pr + 3, v.w);
    atomicAdd(pc + 0, v.x); atomicAdd(pc + 1, v.y);
    atomicAdd(pc + 2, v.z); atomicAdd(pc + 3, v.w);
}

// xb[e] += (a2b[row[e]] + a2b[col[e]]) * 0.5
__global__ void a2b_update(const float* __restrict__ a2b,
                           const long long* __restrict__ row,
                           const long long* __restrict__ col,
                           float* xb, int nE) {
    long long gid = blockIdx.x * (long long)blockDim.x + threadIdx.x;
    long long e = gid >> 5;
    int c = (int)(gid & 31);
    if (e >= nE) return;
    int r = (int)row[e], cl = (int)col[e];
    float4 va = ((const float4*)(a2b + (size_t)r  * DFEAT))[c];
    float4 vb = ((const float4*)(a2b + (size_t)cl * DFEAT))[c];
    float4* p = (float4*)(xb + e * DFEAT) + c;
    float4 v = *p;
    v.x += (va.x + vb.x) * 0.5f;
    v.y += (va.y + vb.y) * 0.5f;
    v.z += (va.z + vb.z) * 0.5f;
    v.w += (va.w + vb.w) * 0.5f;
    *p = v;
}

// ---------------------------------------------------------------------------
// Orchestration
// ---------------------------------------------------------------------------
static inline int cdiv_ll(long long a, long long b) { return (int)((a + b - 1) / b); }

extern "C" void kernel_launch(void* const* d_in, const int* in_sizes, int n_in,
                              void* d_out, int out_size, void* d_ws, size_t ws_size,
                              hipStream_t stream) {
    const float*     x      = (const float*)d_in[0];
    const long long* ei     = (const long long*)d_in[1];   // [2,E] int64
    const long long* lg     = (const long long*)d_in[2];   // [2,E_LG] int64
    const float*     W_atom = (const float*)d_in[3];
    const float*     b_atom = (const float*)d_in[4];
    const float*     W_bond = (const float*)d_in[5];
    const float*     b_bond = (const float*)d_in[6];
    const float*     W_a2b  = (const float*)d_in[7];
    const float*     b_a2b  = (const float*)d_in[8];
    const float*     W_b2a  = (const float*)d_in[9];
    const float*     b_b2a  = (const float*)d_in[10];

    const int N   = in_sizes[0] / DFEAT;
    const int E   = in_sizes[1] / 2;
    const int ELG = in_sizes[2] / 2;

    float* x_atom = (float*)d_out;                         // [N,128]
    float* x_bond = x_atom + (size_t)N * DFEAT;            // [E,128]

    // workspace partition
    float* buf_e = (float*)d_ws;                           // [E,128]
    float* buf_n = buf_e + (size_t)E * DFEAT;              // [N,128]
    float* deg_a = buf_n + (size_t)N * DFEAT;              // [N]
    float* deg_b = deg_a + (((size_t)N + 63) & ~(size_t)63); // [E]

    const long long* ei_row = ei;
    const long long* ei_col = ei + E;
    const long long* lg_src = lg;
    const long long* lg_dst = lg + ELG;

    // ---- 1) atom GCN: xw = x @ W_atom ----
    gemm128_wmma<<<cdiv_ll(N, 64), 128, 0, stream>>>(x, W_atom, nullptr, buf_n, N, 0);
    set_const_f32<<<cdiv_ll(N, 256), 256, 0, stream>>>(deg_a, 1.0f, N);
    deg_count<<<cdiv_ll(E, 256), 256, 0, stream>>>(ei_col, deg_a, E);
    rsqrt_inplace<<<cdiv_ll(N, 256), 256, 0, stream>>>(deg_a, N);
    gcn_self_init<<<cdiv_ll((long long)N * 32, 256), 256, 0, stream>>>(buf_n, deg_a, x_atom, N);
    gcn_scatter<<<cdiv_ll((long long)E * 32, 256), 256, 0, stream>>>(buf_n, ei_row, ei_col, deg_a, x_atom, E);
    bias_relu<<<cdiv_ll((long long)N * 32, 256), 256, 0, stream>>>(x_atom, b_atom, N);

    // ---- 2) bond features = mean of endpoint atoms ----
    bond_mean<<<cdiv_ll((long long)E * 32, 256), 256, 0, stream>>>(x_atom, ei_row, ei_col, buf_e, E);

    // ---- 3) line-graph GCN over bonds (GEMM in place) ----
    gemm128_wmma<<<cdiv_ll(E, 64), 128, 0, stream>>>(buf_e, W_bond, nullptr, buf_e, E, 0);
    set_const_f32<<<cdiv_ll(E, 256), 256, 0, stream>>>(deg_b, 1.0f, E);
    deg_count<<<cdiv_ll(ELG, 256), 256, 0, stream>>>(lg_dst, deg_b, ELG);
    rsqrt_inplace<<<cdiv_ll(E, 256), 256, 0, stream>>>(deg_b, E);
    gcn_self_init<<<cdiv_ll((long long)E * 32, 256), 256, 0, stream>>>(buf_e, deg_b, x_bond, E);
    gcn_scatter<<<cdiv_ll((long long)ELG * 32, 256), 256, 0, stream>>>(buf_e, lg_src, lg_dst, deg_b, x_bond, ELG);
    bias_relu<<<cdiv_ll((long long)E * 32, 256), 256, 0, stream>>>(x_bond, b_bond, E);

    // ---- 4) bond -> atom: b2a = x_bond @ W_b2a + b_b2a, scatter to both ends ----
    gemm128_wmma<<<cdiv_ll(E, 64), 128, 0, stream>>>(x_bond, W_b2a, b_b2a, buf_e, E, 0);
    b2a_scatter<<<cdiv_ll((long long)E * 32, 256), 256, 0, stream>>>(buf_e, ei_row, ei_col, x_atom, E);

    // ---- 5) atom -> bond: a2b = relu(x_atom @ W_a2b + b_a2b), mean of ends ----
    gemm128_wmma<<<cdiv_ll(N, 64), 128, 0, stream>>>(x_atom, W_a2b, b_a2b, buf_n, N, 1);
    a2b_update<<<cdiv_ll((long long)E * 32, 256), 256, 0, stream>>>(buf_n, ei_row, ei_col, x_bond, E);
}